// UnPooling_74474732913173
// MI455X (gfx1250) — compile-verified
//
#include <hip/hip_runtime.h>

// Stride-2 zero-insertion unpool (conv2d_transpose, identity kernel @ top-left).
// Input : [B=32, 112, 112, 64] fp32 NHWC -> Output: [B=32, 224, 224, 64] fp32,
// out[:, ::2, ::2, :] = in, everything else 0.
//
// 0 FLOPs, ~514 MB traffic -> HBM-bound (~22us @ 23.3 TB/s). v4 design:
//  * each thread covers 8 output rows at one (W, c4) position; all accesses are
//    immediate IOFFSETs off one base pointer (row strides fit signed-24-bit).
//  * EVEN output rows (may hold data): predicated NT global_load_b128 +
//    NT global_store_b128.
//  * ODD output rows (always zero, 205 MB total): GLOBAL_STORE_ASYNC_FROM_LDS_B128
//    — the CDNA5 async store engine reads a 16 B zero slot from LDS per lane, so
//    zero-data never occupies VGPR read ports and completion rides ASYNCcnt,
//    decoupled from the data-row load->store chain. Same HBM bytes, less wave work.
//  * v4 fix: LDS offset for the async store is derived from &zbuf[tid] (ptrtoint
//    capture) so the LDS zero-store cannot be dead-store-eliminated. Low 32 bits
//    of the flat address of a __shared__ object ARE the LDS byte offset (ISA §10.2).
//  * every output byte written exactly once; no memset prepass (no double write).

typedef float v4f __attribute__((ext_vector_type(4)));

#define UP_B   32
#define UP_HI  112
#define UP_WI  112
#define UP_C4  16            // 64 channels / 4 floats
#define UP_HO  224
#define UP_WO  224
#define UP_ROWS_PER_THREAD 8
#define UP_ROW4_OUT (UP_WO * UP_C4)   // 3584 float4 per output row
#define UP_ROW4_IN  (UP_WI * UP_C4)   // 1792 float4 per input row

__global__ __launch_bounds__(256)
void UnPooling_74474732913173_kernel(const v4f* __restrict__ in, v4f* __restrict__ out) {
    // One 4 KB block of zeros in LDS (one 16 B slot per thread).
    __shared__ v4f zbuf[256];

    const int tid = threadIdx.x;
    const v4f z = {0.0f, 0.0f, 0.0f, 0.0f};
    zbuf[tid] = z;

    // LDS byte offset of this lane's zero slot, derived FROM THE ADDRESS so the
    // store above is captured and cannot be eliminated. AS3->flat cast puts the
    // LDS offset in the low 32 bits of the flat address.
    const unsigned ldsoff = (unsigned)(unsigned long long)&zbuf[tid];

    // Make the LDS zeros visible to the async store engine (own-slot -> per-wave wait).
    asm volatile("s_wait_dscnt 0x0" ::: "memory");

    // grid = (UP_WO*UP_C4/256 = 14, UP_HO/8 = 28, UP_B = 32)
    const int x  = blockIdx.x * 256 + tid;           // 0 .. 3583 : (W, c4) position
    const int W  = x >> 4;
    const int v  = x & 15;
    const int b  = blockIdx.z;
    const int H0 = blockIdx.y * UP_ROWS_PER_THREAD;  // even by construction

    const bool wEven = (W & 1) == 0;

    // Base pointers; loop accesses use compile-time immediate offsets only.
    v4f* op = out + (size_t)((b * UP_HO + H0) * UP_WO) * UP_C4 + x;
    const v4f* ip = in + (size_t)((b * UP_HI + (H0 >> 1)) * UP_WI + (W >> 1)) * UP_C4 + v;

#pragma unroll
    for (int h = 0; h < UP_ROWS_PER_THREAD; ++h) {
        if ((h & 1) == 0) {
            // Even output row: data at even W, zero at odd W.
            v4f val = z;
            if (wEven) {
                val = __builtin_nontemporal_load(ip + (h >> 1) * UP_ROW4_IN);
            }
            __builtin_nontemporal_store(val, op + h * UP_ROW4_OUT);
        } else {
            // Odd output row: all zeros -> async store engine, data sourced from LDS.
            unsigned long long ga = (unsigned long long)(op + h * UP_ROW4_OUT);
            asm volatile("global_store_async_from_lds_b128 %0, %1, off"
                         :: "v"(ga), "v"(ldsoff)
                         : "memory");
        }
    }

    // Ensure all async zero-row stores completed before wave teardown.
    asm volatile("s_wait_asynccnt 0x0" ::: "memory");
}

extern "C" void kernel_launch(void* const* d_in, const int* in_sizes, int n_in,
                              void* d_out, int out_size, void* d_ws, size_t ws_size,
                              hipStream_t stream) {
    (void)in_sizes; (void)n_in; (void)out_size; (void)d_ws; (void)ws_size;
    const v4f* in = (const v4f*)d_in[0];
    v4f*      out = (v4f*)d_out;

    dim3 block(256, 1, 1);
    dim3 grid((UP_WO * UP_C4) / 256,           // 14
              UP_HO / UP_ROWS_PER_THREAD,      // 28
              UP_B);                           // 32
    UnPooling_74474732913173_kernel<<<grid, block, 0, stream>>>(in, out);
}